// PEQLayer_37495064494840
// MI455X (gfx1250) — compile-verified
//
#include <hip/hip_runtime.h>
#include <math.h>

// ---------------- problem geometry ----------------
#define SRATE   44100.0f
#define BATCH   64
#define TLEN    262144
#define BLK     16
#define NBLK    (TLEN / BLK)        // 16384 blocks per channel
#define NSUB    128                 // blocks per superblock
#define NSUPER  (NBLK / NSUB)       // 128 superblocks per channel
#define TILES_PER_CH (NBLK / 16)    // 1024 WMMA output tiles per channel
#define TPW     4                   // tiles per wave in K2/K5
#define PI_F    3.14159265358979323846f

// ---------------- workspace layout (float units) ----------------
#define CH_STRIDE 1280
#define L_OFF   0      // 16x16 lower-tri Toeplitz impulse matrix        (256)
#define WP_OFF  256    // 16x16 padded input->state map (rows 12..15 =0) (256)
#define M_OFF   512    // 12x12 state transition over one block (A^16)   (144)
#define GP_OFF  656    // 16x16 padded state->output map (cols 12..15=0) (256)
#define MS_OFF  912    // 12x12 superblock transition (A^2048)           (144)
#define COEF_FLOATS (BATCH * CH_STRIDE)
// time-major, 16-row padded: element (block n, state i) at [n*16 + i]
#define DMAT_OFF  COEF_FLOATS
#define DMAT_FLOATS ((size_t)BATCH * NBLK * 16)
#define SMAT_OFF  (DMAT_OFF + DMAT_FLOATS)
#define SMAT_FLOATS ((size_t)BATCH * NBLK * 16)
#define EST_OFF   (SMAT_OFF + SMAT_FLOATS)         // superblock end states
#define EST_FLOATS (BATCH * NSUPER * 12)
#define SENT_OFF  (EST_OFF + EST_FLOATS)           // superblock entry states

typedef __attribute__((ext_vector_type(2))) float v2f;
typedef __attribute__((ext_vector_type(8))) float v8f;

// One step of the 6-stage DF2T cascade (12 states).
__device__ __forceinline__ float casc_step(const float* b0, const float* b1, const float* b2,
                                           const float* a1, const float* a2,
                                           float v, float* z) {
#pragma unroll
  for (int s = 0; s < 6; ++s) {
    float yv  = b0[s] * v + z[2 * s];
    float z0n = b1[s] * v - a1[s] * yv + z[2 * s + 1];
    float z1n = b2[s] * v - a2[s] * yv;
    z[2 * s] = z0n; z[2 * s + 1] = z1n;
    v = yv;
  }
  return v;
}

__device__ __forceinline__ float fmap_dev(float p, float lo, float hi) {
  const float Em1 = 1.7182818284590452354f;       // e - 1
  return lo + (expf(p) - 1.0f) / Em1 * (hi - lo);
}

// ---------------- K1: per-channel coefficient + block-matrix setup ----------------
__global__ __launch_bounds__(64) void peq_setup(const float* __restrict__ yp,
                                                float* __restrict__ ws) {
  int c = threadIdx.x;
  if (c >= BATCH) return;

  float Y[18];
#pragma unroll
  for (int i = 0; i < 18; ++i) {
    float v = yp[c * 18 + i];
    Y[i] = (v != v) ? 0.0f : v;                   // NaN -> 0, as in reference
  }

  float g[6]  = { Y[0]*48.f-24.f, Y[3]*48.f-24.f, Y[6]*48.f-24.f,
                  Y[9]*48.f-24.f, Y[11]*48.f-24.f, Y[15]*48.f-24.f };
  float fc[6] = { fmap_dev(Y[1],16.f,128.f),    fmap_dev(Y[4],128.f,512.f),
                  fmap_dev(Y[7],512.f,1024.f),  fmap_dev(Y[10],1024.f,2048.f),
                  fmap_dev(Y[12],2048.f,4096.f),fmap_dev(Y[16],4096.f,10240.f) };
  float q[6]  = { Y[2]*8.f+0.707f, Y[5]*8.f+0.707f, Y[8]*8.f+0.707f,
                  Y[11]*8.f+0.707f, fabsf(Y[14])*8.f+0.707f, Y[17]*8.f+0.707f };

  const int ft[6] = {1, 2, 2, 2, 2, 0};
  float b0c[6], b1c[6], b2c[6], a1c[6], a2c[6];
#pragma unroll
  for (int s = 0; s < 6; ++s) {
    float A     = powf(10.0f, g[s] / 40.0f);
    float w0    = 2.0f * PI_F * (fc[s] / SRATE);
    float alpha = sinf(w0) / (2.0f * q[s]);
    float cw    = cosf(w0);
    float sA    = sqrtf(A);
    float b0, b1, b2, a0, a1, a2;
    if (ft[s] == 0) {            // high shelf
      b0 =  A * (A + 1 + (A - 1) * cw + 2 * sA * alpha);
      b1 = -2 * A * (A - 1 + (A + 1) * cw);
      b2 =  A * (A + 1 + (A - 1) * cw - 2 * sA * alpha);
      a0 =  A + 1 - (A - 1) * cw + 2 * sA * alpha;
      a1 =  2 * (A - 1 - (A + 1) * cw);
      a2 =  A + 1 - (A - 1) * cw - 2 * sA * alpha;
    } else if (ft[s] == 1) {     // low shelf
      b0 =  A * (A + 1 - (A - 1) * cw + 2 * sA * alpha);
      b1 =  2 * A * (A - 1 - (A + 1) * cw);
      b2 =  A * (A + 1 - (A - 1) * cw - 2 * sA * alpha);
      a0 =  A + 1 + (A - 1) * cw + 2 * sA * alpha;
      a1 = -2 * (A - 1 + (A + 1) * cw);
      a2 =  A + 1 + (A - 1) * cw - 2 * sA * alpha;
    } else {                     // peaking
      b0 = 1 + alpha * A;
      b1 = -2 * cw;
      b2 = 1 - alpha * A;
      a0 = 1 + alpha / A;
      a1 = -2 * cw;
      a2 = 1 - alpha / A;
    }
    b0c[s] = b0 / a0; b1c[s] = b1 / a0; b2c[s] = b2 / a0;
    a1c[s] = a1 / a0; a2c[s] = a2 / a0;
  }

  float* cw_ws = ws + c * CH_STRIDE;

  // zero the padded matrices first
  for (int i = 0; i < 256; ++i) { cw_ws[WP_OFF + i] = 0.0f; cw_ws[GP_OFF + i] = 0.0f; }

  // impulse response h[0..15] + state trajectory -> Wpad columns
  float h[16], z[12];
#pragma unroll
  for (int i = 0; i < 12; ++i) z[i] = 0.f;
  for (int t = 0; t < 16; ++t) {
    float v = casc_step(b0c, b1c, b2c, a1c, a2c, (t == 0) ? 1.0f : 0.0f, z);
    h[t] = v;
    // impulse at in-block position j reaches block end after 16-j steps:
    // Wpad[i][15-t] = state i after (t+1) steps of an impulse at step 0
    for (int i = 0; i < 12; ++i) cw_ws[WP_OFF + i * 16 + (15 - t)] = z[i];
  }
  // L (16x16 lower-triangular Toeplitz of h)
  for (int m = 0; m < 16; ++m)
    for (int k = 0; k < 16; ++k)
      cw_ws[L_OFF + m * 16 + k] = (m >= k) ? h[m - k] : 0.0f;

  // Gpad (16x16, cols 12..15 zero) and M (12x12)
  for (int i = 0; i < 12; ++i) {
#pragma unroll
    for (int r = 0; r < 12; ++r) z[r] = 0.f;
    z[i] = 1.0f;
    for (int t = 0; t < 16; ++t) {
      float v = casc_step(b0c, b1c, b2c, a1c, a2c, 0.0f, z);
      cw_ws[GP_OFF + t * 16 + i] = v;
    }
    for (int r = 0; r < 12; ++r) cw_ws[M_OFF + r * 12 + i] = z[r];
  }

  // M_super = M^NSUB via 7 squarings (NSUB = 128)
  float P[144], Q[144];
  for (int i = 0; i < 144; ++i) P[i] = cw_ws[M_OFF + i];
  for (int it = 0; it < 7; ++it) {
    for (int i = 0; i < 12; ++i)
      for (int j = 0; j < 12; ++j) {
        float acc = 0.f;
        for (int k = 0; k < 12; ++k) acc += P[i * 12 + k] * P[k * 12 + j];
        Q[i * 12 + j] = acc;
      }
    for (int i = 0; i < 144; ++i) P[i] = Q[i];
  }
  for (int i = 0; i < 144; ++i) cw_ws[MS_OFF + i] = P[i];
}

// ---------------- K2: dual GEMM per tile via fp32 WMMA ----------------
// Yp(16 x NBLK) = L(16x16)    * X(16 x NBLK)   -> out      [n*16 + t]
// D (16 x NBLK) = Wpad(16x16) * X(16 x NBLK)   -> Dmat     [n*16 + i]  (rows 12..15 = 0)
// One wave handles TPW tiles; A fragments loaded once, B fragments reused by both GEMMs.
// Both result tiles store as branch-free 2x global_store_b128.
__global__ __launch_bounds__(256) void peq_fir_wmma(const float* __restrict__ x,
                                                    const float* __restrict__ ws,
                                                    float* __restrict__ out,
                                                    float* __restrict__ Dmat) {
  int wave = (blockIdx.x * blockDim.x + threadIdx.x) >> 5;
  int lane = threadIdx.x & 31;
  int c     = wave >> 8;                 // 256 waves per channel
  int tbase = (wave & 255) * TPW;

  const float* L  = ws + c * CH_STRIDE + L_OFF;
  const float* Wp = ws + c * CH_STRIDE + WP_OFF;
  const float* xc = x + (size_t)c * TLEN;
  float*       oc = out + (size_t)c * TLEN;
  float*       Dc = Dmat + (size_t)c * NBLK * 16;

  int m = lane & 15, half = lane >> 4;

  v2f aL[4], aW[4];
#pragma unroll
  for (int kk = 0; kk < 4; ++kk) {
    int k0 = kk * 4 + half * 2;
    aL[kk].x = L[m * 16 + k0];  aL[kk].y = L[m * 16 + k0 + 1];
    aW[kk].x = Wp[m * 16 + k0]; aW[kk].y = Wp[m * 16 + k0 + 1];
  }

  for (int t = 0; t < TPW; ++t) {
    int tile = tbase + t;
    int n    = tile * 16 + m;            // block/column index
    v8f accY = {}, accD = {};
#pragma unroll
    for (int kk = 0; kk < 4; ++kk) {
      int k0 = kk * 4 + half * 2;
      const float* xb = xc + (size_t)n * 16 + k0;
      v2f b; b.x = xb[0]; b.y = xb[1];
      accY = __builtin_amdgcn_wmma_f32_16x16x4_f32(false, aL[kk], false, b,
                                                   (short)0, accY, false, false);
      accD = __builtin_amdgcn_wmma_f32_16x16x4_f32(false, aW[kk], false, b,
                                                   (short)0, accD, false, false);
    }
    // D layout: VGPR r, lanes 0-15 -> (row r, col n); lanes 16-31 -> (row r+8, col n)
    float* op = oc + (size_t)n * 16 + half * 8;
    float* dp = Dc + (size_t)n * 16 + half * 8;
#pragma unroll
    for (int r = 0; r < 8; ++r) op[r] = accY[r];
#pragma unroll
    for (int r = 0; r < 8; ++r) dp[r] = accD[r];
  }
}

// ---------------- K3: lane-parallel affine scan over blocks ----------------
// Lane i (i<12) owns state row i; its 12x12 M row lives in registers; d_i streams
// from time-major Dmat (12 contiguous floats per block). Cross-lane state
// broadcast via shfl within 16-lane halves; each wave runs two superblocks.
// phase 0: zero entry state, emit superblock end states.
// phase 1: true entry state from Sent, emit per-block entry states to Smat.
__global__ __launch_bounds__(256) void peq_scan(const float* __restrict__ ws,
                                                const float* __restrict__ Dmat,
                                                float* __restrict__ Smat,
                                                float* __restrict__ Est,
                                                const float* __restrict__ Sent,
                                                int phase) {
  int wave = (blockIdx.x * blockDim.x + threadIdx.x) >> 5;
  int lane = threadIdx.x & 31;
  int c    = wave >> 6;                  // 64 waves per channel
  int widx = wave & 63;
  int i    = lane & 15;
  int ii   = (i < 12) ? i : 11;          // clamp for safe loads on pad lanes
  int sb   = widx * 2 + (lane >> 4);     // two superblocks per wave

  const float* Mw = ws + c * CH_STRIDE + M_OFF + ii * 12;
  float Mrow[12];
#pragma unroll
  for (int j = 0; j < 12; ++j) Mrow[j] = Mw[j];

  const float* Dp = Dmat + ((size_t)c * NBLK + (size_t)sb * NSUB) * 16 + ii;
  float*       Sp = Smat + ((size_t)c * NBLK + (size_t)sb * NSUB) * 16 + i;

  float s = phase ? Sent[(c * NSUPER + sb) * 12 + ii] : 0.0f;

  for (int blk = 0; blk < NSUB; ++blk) {
    float d = Dp[(size_t)blk * 16];
    if (phase && i < 12) Sp[(size_t)blk * 16] = s;   // record block entry state
    float ns = d;
#pragma unroll
    for (int j = 0; j < 12; ++j) ns += Mrow[j] * __shfl(s, j, 16);
    s = ns;
  }
  if (!phase && i < 12) Est[(c * NSUPER + sb) * 12 + i] = s;
}

// ---------------- K4: 128-step scan over superblocks (per channel) ----------------
__global__ __launch_bounds__(64) void peq_superscan(const float* __restrict__ ws,
                                                    float* __restrict__ Sent,
                                                    const float* __restrict__ Est) {
  int c = threadIdx.x;
  if (c >= BATCH) return;
  const float* Ms = ws + c * CH_STRIDE + MS_OFF;
  float S[12];
#pragma unroll
  for (int i = 0; i < 12; ++i) S[i] = 0.0f;
  for (int j = 0; j < NSUPER; ++j) {
#pragma unroll
    for (int i = 0; i < 12; ++i) Sent[(c * NSUPER + j) * 12 + i] = S[i];
    float ns[12];
#pragma unroll
    for (int i = 0; i < 12; ++i) {
      float acc = 0.f;
#pragma unroll
      for (int k = 0; k < 12; ++k) acc += Ms[i * 12 + k] * S[k];
      ns[i] = acc + Est[(c * NSUPER + j) * 12 + i];
    }
#pragma unroll
    for (int i = 0; i < 12; ++i) S[i] = ns[i];
  }
}

// ---------------- K5: homogeneous fix-up GEMM + finite mask ----------------
// out += Gpad(16x12) * S(12 x NBLK); K=12 -> 3 WMMAs per tile; the existing out
// tile is the C accumulator; B fragments are contiguous b64 loads from
// time-major Smat; reference's non-finite -> 0 mask applied on final store.
__global__ __launch_bounds__(256) void peq_fix_wmma(const float* __restrict__ ws,
                                                    const float* __restrict__ Smat,
                                                    float* __restrict__ out) {
  int wave = (blockIdx.x * blockDim.x + threadIdx.x) >> 5;
  int lane = threadIdx.x & 31;
  int c     = wave >> 8;                 // 256 waves per channel
  int tbase = (wave & 255) * TPW;

  const float* Gp = ws + c * CH_STRIDE + GP_OFF;
  const float* Sc = Smat + (size_t)c * NBLK * 16;
  float*       oc = out + (size_t)c * TLEN;

  int m = lane & 15, half = lane >> 4;

  v2f aG[3];
#pragma unroll
  for (int kk = 0; kk < 3; ++kk) {
    int k0 = kk * 4 + half * 2;
    aG[kk].x = Gp[m * 16 + k0]; aG[kk].y = Gp[m * 16 + k0 + 1];
  }

  for (int t = 0; t < TPW; ++t) {
    int tile = tbase + t;
    int n    = tile * 16 + m;

    float* op = oc + (size_t)n * 16 + half * 8;
    v8f acc;
#pragma unroll
    for (int r = 0; r < 8; ++r) acc[r] = op[r];  // C = current zero-state output

#pragma unroll
    for (int kk = 0; kk < 3; ++kk) {             // K = 12 states
      int k0 = kk * 4 + half * 2;
      const float* sp = Sc + (size_t)n * 16 + k0;
      v2f b; b.x = sp[0]; b.y = sp[1];
      acc = __builtin_amdgcn_wmma_f32_16x16x4_f32(false, aG[kk], false, b,
                                                  (short)0, acc, false, false);
    }
#pragma unroll
    for (int r = 0; r < 8; ++r) {
      float v = acc[r];
      op[r] = isfinite(v) ? v : 0.0f;
    }
  }
}

// ---------------- host entry ----------------
extern "C" void kernel_launch(void* const* d_in, const int* in_sizes, int n_in,
                              void* d_out, int out_size, void* d_ws, size_t ws_size,
                              hipStream_t stream) {
  const float* x = (const float*)d_in[0];   // [64, 262144]
  const float* y = (const float*)d_in[1];   // [64, 18]
  float* out  = (float*)d_out;              // [64, 262144]
  float* ws   = (float*)d_ws;
  float* Dmat = ws + DMAT_OFF;
  float* Smat = ws + SMAT_OFF;
  float* Est  = ws + EST_OFF;
  float* Sent = ws + SENT_OFF;

  peq_setup<<<1, 64, 0, stream>>>(y, ws);
  // K2: 64 ch * 256 waves = 16384 waves, 8 waves/block
  peq_fir_wmma<<<2048, 256, 0, stream>>>(x, ws, out, Dmat);
  // K3 phase 0: 64 ch * 64 waves = 4096 waves
  peq_scan<<<512, 256, 0, stream>>>(ws, Dmat, Smat, Est, Sent, 0);
  peq_superscan<<<1, 64, 0, stream>>>(ws, Sent, Est);
  // K3 phase 1
  peq_scan<<<512, 256, 0, stream>>>(ws, Dmat, Smat, Est, Sent, 1);
  // K5: 64 ch * 256 waves = 16384 waves, 8 waves/block
  peq_fix_wmma<<<2048, 256, 0, stream>>>(ws, Smat, out);
}